// RateConv_89008902242858
// MI455X (gfx1250) — compile-verified
//
#include <hip/hip_runtime.h>
#include <hip/hip_bf16.h>

#define N_NODES 100000
#define N_EDGES 1600000
#define DFEAT   128
#define N_RATES 5

typedef __attribute__((ext_vector_type(2))) float v2f;
typedef __attribute__((ext_vector_type(8))) float v8f;

__device__ __forceinline__ float rsqrt_clamp1(float x) {
    // deg^{-1/2} with dgl's clamp deg>=1; v_rsq_f32 (~1ulp) on exact integers
    return __builtin_amdgcn_rsqf(fmaxf(x, 1.0f));
}

// ---------------------------------------------------------------------------
// Kernel 1: per-rate degree counts (fp32 atomics; degrees << 2^24 so exact)
// ---------------------------------------------------------------------------
__global__ void deg_kernel(const int* __restrict__ src,
                           const int* __restrict__ dst,
                           const int* __restrict__ rate,
                           float* __restrict__ out_deg,
                           float* __restrict__ in_deg) {
    int e = blockIdx.x * blockDim.x + threadIdx.x;
    if (e >= N_EDGES) return;
    int r = rate[e];
    atomicAdd(&out_deg[(size_t)r * N_NODES + src[e]], 1.0f);
    atomicAdd(&in_deg [(size_t)r * N_NODES + dst[e]], 1.0f);
}

// ---------------------------------------------------------------------------
// Kernel 2: edge scatter: agg[r][dst] += h[src] * out_deg[r][src]^-1/2
// One wave32 per edge, 4 floats per lane (float4 gather, 4 fp32 atomics).
// h is 51MB -> L2-resident gather; atomics land mostly in 192MB L2.
// ---------------------------------------------------------------------------
__global__ void scatter_kernel(const float* __restrict__ h,
                               const int* __restrict__ src,
                               const int* __restrict__ dst,
                               const int* __restrict__ rate,
                               const float* __restrict__ out_deg,
                               float* __restrict__ agg) {
    long long tid = (long long)blockIdx.x * blockDim.x + threadIdx.x;
    int e    = (int)(tid >> 5);
    int lane = (int)(tid & 31);
    if (e >= N_EDGES) return;
    int s = src[e];
    int d = dst[e];
    int r = rate[e];
    float ns = rsqrt_clamp1(out_deg[(size_t)r * N_NODES + s]);
    const float4 hv = *(const float4*)(h + (size_t)s * DFEAT + lane * 4);
    float* ap = agg + ((size_t)r * N_NODES + d) * DFEAT + lane * 4;
    atomicAdd(ap + 0, hv.x * ns);
    atomicAdd(ap + 1, hv.y * ns);
    atomicAdd(ap + 2, hv.z * ns);
    atomicAdd(ap + 3, hv.w * ns);
}

// ---------------------------------------------------------------------------
// Kernel 3: out[m, r*128+n] = (agg[r] @ W[r])[m,n] * in_deg[r][m]^-1/2 + b[r][n]
// Block = 256 threads = 8 waves; covers one (16-row m-tile, rate).
// Wave w computes the 16x16 tile at n0 = 16*w with V_WMMA_F32_16X16X4_F32
// over K=128 (32 WMMAs, exact fp32 accumulation).
// A tile staged once in LDS (pad 132 -> conflict-free b64 reads), reused 8x.
// ---------------------------------------------------------------------------
__global__ void gemm_kernel(const float* __restrict__ agg,
                            const float* __restrict__ W,
                            const float* __restrict__ bias,
                            const float* __restrict__ in_deg,
                            float* __restrict__ out) {
    __shared__ float As[16][132];

    const int r    = blockIdx.y;
    const int m0   = blockIdx.x * 16;
    const int t    = threadIdx.x;
    const int wave = t >> 5;
    const int lane = t & 31;
    const int half = lane >> 4;     // 0: lanes 0-15, 1: lanes 16-31
    const int l15  = lane & 15;
    const int n0   = wave << 4;

    // --- stage A tile (16 x 128) into LDS, fully coalesced ---
    {
        int row = t >> 4;           // 16 threads per row
        int kk  = (t & 15) << 3;    // 8 consecutive floats per thread
        const float* g = agg + ((size_t)r * N_NODES + m0 + row) * DFEAT + kk;
        float4 x0 = *(const float4*)(g);
        float4 x1 = *(const float4*)(g + 4);
        *(float4*)&As[row][kk]     = x0;
        *(float4*)&As[row][kk + 4] = x1;
    }
    __syncthreads();

    // --- WMMA main loop: D = A(16x128) x B(128x16) ---
    // A frag (16x4 f32): lanes 0-15 -> M=lane, V0=K0,V1=K1; lanes 16-31 -> K2,K3
    // B frag (4x16 f32): lanes 0-15 -> N=lane, V0=K0,V1=K1; lanes 16-31 -> K2,K3
    const float* Bw = W + (size_t)r * DFEAT * DFEAT + n0 + l15;  // column base
    v8f c = {};
#pragma unroll 8
    for (int k0 = 0; k0 < DFEAT; k0 += 4) {
        int kb = k0 + (half << 1);
        v2f a = *(const v2f*)&As[l15][kb];          // A[m][kb], A[m][kb+1]
        v2f bv;
        bv.x = Bw[(size_t)kb * DFEAT];              // W[kb][n]
        bv.y = Bw[(size_t)(kb + 1) * DFEAT];        // W[kb+1][n]
        c = __builtin_amdgcn_wmma_f32_16x16x4_f32(
                /*neg_a=*/false, a, /*neg_b=*/false, bv,
                /*c_mod=*/(short)0, c, /*reuse_a=*/false, /*reuse_b=*/false);
    }

    // --- epilogue: dst-norm + bias; C/D layout: VGPR v -> M = v + 8*half ---
    const float bval = bias[(size_t)r * DFEAT + n0 + l15];
    const float* idg = in_deg + (size_t)r * N_NODES;
#pragma unroll
    for (int v = 0; v < 8; ++v) {
        int row  = m0 + v + (half << 3);
        float nd = rsqrt_clamp1(idg[row]);
        out[(size_t)row * (N_RATES * DFEAT) + r * DFEAT + n0 + l15] =
            c[v] * nd + bval;
    }
}

// ---------------------------------------------------------------------------
// Launcher. d_ws layout:
//   [0, 2MB)      out_deg  : 5 * 100000 f32
//   [2MB, 4MB)    in_deg   : 5 * 100000 f32
//   [4MB, 260MB)  agg      : 5 * 100000 * 128 f32
// ---------------------------------------------------------------------------
extern "C" void kernel_launch(void* const* d_in, const int* in_sizes, int n_in,
                              void* d_out, int out_size, void* d_ws, size_t ws_size,
                              hipStream_t stream) {
    const float* h    = (const float*)d_in[0];
    const float* W    = (const float*)d_in[1];
    const float* b    = (const float*)d_in[2];
    const int*   src  = (const int*)d_in[3];
    const int*   dst  = (const int*)d_in[4];
    const int*   rate = (const int*)d_in[5];
    float* out = (float*)d_out;

    const size_t DEG_FLOATS = (size_t)N_RATES * N_NODES;          // 500,000
    float* out_deg = (float*)d_ws;
    float* in_deg  = out_deg + DEG_FLOATS;
    float* agg     = in_deg + DEG_FLOATS;
    const size_t ZERO_BYTES = (2 * DEG_FLOATS +
                               (size_t)N_RATES * N_NODES * DFEAT) * sizeof(float);

    hipMemsetAsync(d_ws, 0, ZERO_BYTES, stream);

    // degrees: 1 thread / edge
    deg_kernel<<<(N_EDGES + 255) / 256, 256, 0, stream>>>(src, dst, rate,
                                                          out_deg, in_deg);

    // scatter: 1 wave / edge
    {
        long long threads = (long long)N_EDGES * 32;
        int blocks = (int)((threads + 255) / 256);
        scatter_kernel<<<blocks, 256, 0, stream>>>(h, src, dst, rate,
                                                   out_deg, agg);
    }

    // WMMA GEMM + epilogue: grid (6250 m-tiles, 5 rates), 8 waves per block
    dim3 grid(N_NODES / 16, N_RATES, 1);
    gemm_kernel<<<grid, 256, 0, stream>>>(agg, W, b, in_deg, out);
}